// DeformConv2d_55379308315253
// MI455X (gfx1250) — compile-verified
//
#include <hip/hip_runtime.h>
#include <cstdint>

typedef __attribute__((ext_vector_type(2))) float v2f;
typedef __attribute__((ext_vector_type(8))) float v8f;

#define B_ 16
#define CIN 64
#define COUT 64
#define HW 64          // H == W == 64
#define KK 9           // 3x3 taps
#define OFFC 18        // 2*KK offset channels
#define KTOT (CIN * KK)      // 576 GEMM K
#define CCH 16               // channels per K-chunk
#define KCH (CCH * KK)       // 144 rows per chunk
#define PSTRIDE 160          // floats per K-pair row: 2*64 cols + 32 pad (==32 mod 64)

// ---------------------------------------------------------------------------
// Kernel A: plain 3x3 conv, 64 -> 18 channels (the offset predictor).
// One thread per (b, oy, ox); all 18 output channels accumulated in registers.
// Weight indices are wave-uniform -> scalar loads via constant cache.
// ---------------------------------------------------------------------------
__global__ __launch_bounds__(256) void offset_conv_kernel(
    const float* __restrict__ x, const float* __restrict__ w_off,
    const float* __restrict__ b_off, float* __restrict__ offs) {
  const int idx = blockIdx.x * 256 + threadIdx.x;      // 16*64*64 = 65536 total
  const int b  = idx >> 12;
  const int r  = idx & 4095;
  const int oy = r >> 6;
  const int ox = r & 63;

  float acc[OFFC];
#pragma unroll
  for (int c = 0; c < OFFC; ++c) acc[c] = b_off[c];

  for (int ci = 0; ci < CIN; ++ci) {
    const float* xp = x + (((size_t)b * CIN + ci) << 12);
    float xv[KK];
#pragma unroll
    for (int ky = 0; ky < 3; ++ky) {
      const int y = oy - 1 + ky;
      const bool vy = (y >= 0) & (y < HW);
#pragma unroll
      for (int kx = 0; kx < 3; ++kx) {
        const int xx = ox - 1 + kx;
        const bool v = vy & (xx >= 0) & (xx < HW);
        xv[ky * 3 + kx] = v ? xp[y * HW + xx] : 0.0f;
      }
    }
    const float* wp = w_off + ci * KK;     // w_off[c][ci][ky][kx] = [c*576 + ci*9 + t]
#pragma unroll
    for (int c = 0; c < OFFC; ++c) {
      const float* wc = wp + c * KTOT;     // uniform address -> s_load
#pragma unroll
      for (int t = 0; t < KK; ++t) acc[c] = fmaf(xv[t], wc[t], acc[c]);
    }
  }

#pragma unroll
  for (int c = 0; c < OFFC; ++c)
    offs[(((size_t)b * OFFC + c) * HW + oy) * HW + ox] = acc[c];
}

// ---------------------------------------------------------------------------
// Kernel B: deformable conv. One workgroup per (b, oy): 64 pixels x 64 Cout.
// Stage 1: bilinear-gather a [144 x 64] fp32 cols chunk (16 input channels)
//          into LDS, K-pair interleaved so a B fragment is one ds_load_b64.
// Stage 2: 8 waves x 2 WMMA tiles, v_wmma_f32_16x16x4_f32, K stepped by 4,
//          fp32 accumulators live across all 4 chunks.
// ---------------------------------------------------------------------------
__global__ __launch_bounds__(256) void deform_wmma_kernel(
    const float* __restrict__ x, const float* __restrict__ offs,
    const float* __restrict__ w_dcn, const float* __restrict__ b_dcn,
    float* __restrict__ out) {
  // element (r, n) lives at cols[(r>>1)*PSTRIDE + 2n + (r&1)]
  __shared__ __align__(16) float cols[(KCH / 2) * PSTRIDE];   // 46080 B

  const int tid = threadIdx.x;
  const int oy  = blockIdx.x;
  const int b   = blockIdx.y;

  // ---- per-(tap,pixel) bilinear prep, kept in registers across chunks ----
  int   pkk[3], pn[3];
  int   a00[3], a01[3], a10[3], a11[3];
  float w00[3], w01[3], w10[3], w11[3];
  int npair = 0;
  for (int p = tid; p < KK * HW; p += 256, ++npair) {
    const int kk = p >> 6;          // tap 0..8
    const int n  = p & 63;          // output pixel (ox)
    pkk[npair] = kk;
    pn[npair]  = n;
    const float offy = offs[(((size_t)b * OFFC + kk * 2 + 0) * HW + oy) * HW + n];
    const float offx = offs[(((size_t)b * OFFC + kk * 2 + 1) * HW + oy) * HW + n];
    const float py = offy + (float)(kk / 3) + (float)oy - 1.0f;
    const float px = offx + (float)(kk % 3) + (float)n  - 1.0f;
    const float y0f = floorf(py), x0f = floorf(px);
    const float wy1 = py - y0f,   wx1 = px - x0f;
    const float wy0 = 1.0f - wy1, wx0 = 1.0f - wx1;
    const int y0 = (int)y0f, x0 = (int)x0f;
    const int y1 = y0 + 1,   x1 = x0 + 1;
    const float vy0 = (y0 >= 0 && y0 < HW) ? 1.0f : 0.0f;
    const float vy1 = (y1 >= 0 && y1 < HW) ? 1.0f : 0.0f;
    const float vx0 = (x0 >= 0 && x0 < HW) ? 1.0f : 0.0f;
    const float vx1 = (x1 >= 0 && x1 < HW) ? 1.0f : 0.0f;
    w00[npair] = wy0 * wx0 * vy0 * vx0;
    w01[npair] = wy0 * wx1 * vy0 * vx1;
    w10[npair] = wy1 * wx0 * vy1 * vx0;
    w11[npair] = wy1 * wx1 * vy1 * vx1;
    const int yc0 = min(max(y0, 0), HW - 1), yc1 = min(max(y1, 0), HW - 1);
    const int xc0 = min(max(x0, 0), HW - 1), xc1 = min(max(x1, 0), HW - 1);
    a00[npair] = yc0 * HW + xc0;
    a01[npair] = yc0 * HW + xc1;
    a10[npair] = yc1 * HW + xc0;
    a11[npair] = yc1 * HW + xc1;
  }

  // ---- wave / tile assignment (wave32) ----
  const int wv   = tid >> 5;
  const int lane = tid & 31;
  const int lm   = lane & 15;
  const int kof  = (lane >> 4) * 2;          // half-wave K offset (A frag)
  const int prof = kof >> 1;                 // half-wave K-pair offset (B frag)
  const int m0   = (wv >> 1) * 16;           // Cout tile row
  const int n0a  = (wv & 1) * 32;            // two adjacent N tiles share A
  const int n0b  = n0a + 16;
  const int c2a  = 2 * (n0a + lm);
  const int c2b  = 2 * (n0b + lm);

  v8f c0 = {};
  v8f c1 = {};

  for (int cc = 0; cc < CIN / CCH; ++cc) {
    // ---- stage 1: build cols chunk (rows r = ci_local*9 + kk) ----
    for (int i = 0; i < npair; ++i) {
      const int kk = pkk[i], n = pn[i];
      const float* xp = x + (((size_t)b * CIN + cc * CCH) << 12);
#pragma unroll 4
      for (int cl = 0; cl < CCH; ++cl) {
        const float* xc = xp + ((size_t)cl << 12);
        const float v = w00[i] * xc[a00[i]] + w01[i] * xc[a01[i]] +
                        w10[i] * xc[a10[i]] + w11[i] * xc[a11[i]];
        const int r = cl * KK + kk;
        cols[(r >> 1) * PSTRIDE + 2 * n + (r & 1)] = v;
      }
    }
    __syncthreads();

    // ---- stage 2: fp32 WMMA GEMM over this K chunk ----
    const float* Arow = w_dcn + (m0 + lm) * KTOT + cc * KCH + kof;
    __builtin_prefetch((const void*)(Arow + KCH), 0, 0);   // next chunk weights
    for (int ks = 0; ks < KCH / 4; ++ks) {
      v2f a;
      a.x = Arow[ks * 4];
      a.y = Arow[ks * 4 + 1];
      const float* prow = cols + (ks * 2 + prof) * PSTRIDE;
      const v2f bb0 = *(const v2f*)(prow + c2a);   // one ds_load_b64
      const v2f bb1 = *(const v2f*)(prow + c2b);   // one ds_load_b64
      c0 = __builtin_amdgcn_wmma_f32_16x16x4_f32(
          false, a, false, bb0, (short)0, c0, false, false);
      c1 = __builtin_amdgcn_wmma_f32_16x16x4_f32(
          false, a, false, bb1, (short)0, c1, false, false);
    }
    __syncthreads();
  }

  // ---- store D (+bias): VGPR v -> M = m0 + v + 8*(lane>=16), N = n0 + lm ----
  const int mbase = m0 + (lane >> 4) * 8;
#pragma unroll
  for (int v = 0; v < 8; ++v) {
    const int m = mbase + v;
    const float bias = b_dcn[m];
    const size_t o = (((size_t)b * COUT + m) * HW + oy) * HW;
    out[o + n0a + lm] = c0[v] + bias;
    out[o + n0b + lm] = c1[v] + bias;
  }
}

// ---------------------------------------------------------------------------
extern "C" void kernel_launch(void* const* d_in, const int* in_sizes, int n_in,
                              void* d_out, int out_size, void* d_ws, size_t ws_size,
                              hipStream_t stream) {
  const float* x     = (const float*)d_in[0];
  const float* w_off = (const float*)d_in[1];
  const float* b_off = (const float*)d_in[2];
  const float* w_dcn = (const float*)d_in[3];
  const float* b_dcn = (const float*)d_in[4];
  float* out  = (float*)d_out;
  float* offs = (float*)d_ws;   // 16*18*64*64 floats = 4.7 MB scratch

  offset_conv_kernel<<<(B_ * HW * HW) / 256, 256, 0, stream>>>(x, w_off, b_off, offs);
  deform_wmma_kernel<<<dim3(HW, B_), 256, 0, stream>>>(x, offs, w_dcn, b_dcn, out);
}